// SpGraphAttentionLayer_31842887532864
// MI455X (gfx1250) — compile-verified
//
#include <hip/hip_runtime.h>
#include <hip/hip_bf16.h>

// GAT layer on MI455X (gfx1250, wave32).
// B=4, N=10000, E=320000, DIN=DOUT=128, ALPHA=0.2
//
// Phase 1: h = x@W via V_WMMA_F32_16X16X4_F32 (fp32 WMMA, bit-faithful to ref).
//          W staged TRANSPOSED in LDS with padded stride so each B-fragment is
//          a single bank-conflict-free ds_load_b64 into an even VGPR pair.
// Phase 2: f_src/f_dst row dots (wave-per-row, shfl reduce)
// Phase 3: per-edge attention + f32-atomic segment-sum (bandwidth bound ~1.4GB
//          -> ~60us at 23.3 TB/s; agg (20.5MB) lives in the 192MB L2)
// Phase 4: normalize + ELU

#define NB 4
#define NN 10000
#define NE 320000
#define DD 128
#define NROWS (NB * NN)            // 40000 flattened (b,n) rows
#define ALPHA 0.2f
#define WT_STRIDE 130              // padded col stride: bank stride 2/col -> conflict-free

typedef __attribute__((ext_vector_type(2))) float v2f;
typedef __attribute__((ext_vector_type(8))) float v8f;

// ---------------------------------------------------------------------------
// Phase 0: zero agg / rowsum / f scratch (harness poisons d_ws with 0xAA)
// ---------------------------------------------------------------------------
__global__ void gat_zero(float* __restrict__ p, int n) {
  int i = blockIdx.x * blockDim.x + threadIdx.x;
  if (i < n) p[i] = 0.0f;
}

// ---------------------------------------------------------------------------
// Phase 1: h[r, :] = x[r, :] @ W  for r in [0, 40000), fp32 WMMA 16x16x4.
// One wave -> 16 rows x 128 cols. Block = 128 threads = 4 waves.
// ---------------------------------------------------------------------------
__global__ void __launch_bounds__(128)
gat_gemm_wmma(const float* __restrict__ x,
              const float* __restrict__ W,
              float* __restrict__ h) {
  __shared__ float WldsT[DD * WT_STRIDE];   // 66.6 KB of the 320 KB WGP LDS
  const int tid = threadIdx.x;

  // cooperative transpose: WldsT[c][k] = W[k][c]; coalesced global reads
  for (int i = tid; i < DD * DD; i += 128) {
    const int k = i >> 7;
    const int c = i & (DD - 1);
    WldsT[c * WT_STRIDE + k] = W[i];
  }
  __syncthreads();

  const int wave  = tid >> 5;
  const int lane  = tid & 31;
  const int half  = lane >> 4;              // 0: lanes 0-15, 1: lanes 16-31
  const int sub   = lane & 15;
  const int mtile = blockIdx.x * 4 + wave;  // 0..2499, exact (no partial waves)
  const int rbase = mtile * 16;

  v8f acc[8] = {};                          // 8 N-tiles of 16 cols -> 64 VGPRs

  const float* xrow = x + (size_t)(rbase + sub) * DD;  // A row m = sub

  for (int kk = 0; kk < DD; kk += 4) {
    const int k0 = kk + half * 2;           // A/B 16x4 layout: VGPR0 K={0,2}, VGPR1 K={1,3}
    const v2f a = *(const v2f*)(xrow + k0);              // global_load_b64
#pragma unroll
    for (int t = 0; t < 8; ++t) {
      const v2f b = *(const v2f*)(&WldsT[(t * 16 + sub) * WT_STRIDE + k0]); // ds_load_b64
      acc[t] = __builtin_amdgcn_wmma_f32_16x16x4_f32(
          /*neg_a=*/false, a, /*neg_b=*/false, b,
          /*c_mod=*/(short)0, acc[t], /*reuse_a=*/false, /*reuse_b=*/false);
    }
  }

  // C/D layout: VGPR r holds row (r + half*8), col = sub within tile.
#pragma unroll
  for (int t = 0; t < 8; ++t) {
#pragma unroll
    for (int r = 0; r < 8; ++r) {
      const int row = rbase + r + half * 8;
      const int col = t * 16 + sub;
      float v = acc[t][r];
      v = (v != v) ? 0.0f : v;              // NaN -> 0 (matches reference)
      h[(size_t)row * DD + col] = v;
    }
  }
}

// ---------------------------------------------------------------------------
// Phase 2: f_src[r] = h[r,:]·a[0:128], f_dst[r] = h[r,:]·a[128:256].
// One wave32 per row; float4 per lane; shfl_xor tree reduce.
// ---------------------------------------------------------------------------
__global__ void __launch_bounds__(256)
gat_attn_coef(const float* __restrict__ h,
              const float* __restrict__ a,
              float* __restrict__ fsrc,
              float* __restrict__ fdst) {
  const int lane = threadIdx.x & 31;
  const int row  = blockIdx.x * (blockDim.x >> 5) + (threadIdx.x >> 5);
  if (row >= NROWS) return;

  const float4 hv = ((const float4*)(h + (size_t)row * DD))[lane];
  const float4 as = ((const float4*)a)[lane];
  const float4 ad = ((const float4*)(a + DD))[lane];

  float ss = hv.x * as.x + hv.y * as.y + hv.z * as.z + hv.w * as.w;
  float sd = hv.x * ad.x + hv.y * ad.y + hv.z * ad.z + hv.w * ad.w;
#pragma unroll
  for (int off = 16; off > 0; off >>= 1) {
    ss += __shfl_xor(ss, off, 32);
    sd += __shfl_xor(sd, off, 32);
  }
  if (lane == 0) { fsrc[row] = ss; fdst[row] = sd; }
}

// ---------------------------------------------------------------------------
// Phase 3: per-edge.  e = exp(-leaky_relu(f_src[src]+f_dst[dst]))
//   rowsum[src] += e ;  agg[src,:] += e * h[dst,:]
// One wave32 per edge, all 4 batches; float4 gather, f32 atomics resolve in L2.
// ---------------------------------------------------------------------------
__global__ void __launch_bounds__(256)
gat_edge(const int* __restrict__ ei,
         const float* __restrict__ h,
         const float* __restrict__ fsrc,
         const float* __restrict__ fdst,
         float* __restrict__ agg,
         float* __restrict__ rowsum) {
  const int lane = threadIdx.x & 31;
  const int e = blockIdx.x * (blockDim.x >> 5) + (threadIdx.x >> 5);
  if (e >= NE) return;

  const int src = ei[e];
  const int dst = ei[NE + e];

#pragma unroll
  for (int b = 0; b < NB; ++b) {
    const int rs = b * NN + src;
    const int rd = b * NN + dst;
    const float s  = fsrc[rs] + fdst[rd];
    const float lr = (s > 0.0f) ? s : ALPHA * s;
    float ew = __expf(-lr);
    ew = (ew != ew) ? 0.0f : ew;

    if (lane == 0) atomicAdd(rowsum + rs, ew);

    const float4 hv = ((const float4*)(h + (size_t)rd * DD))[lane];
    float* ag = agg + (size_t)rs * DD + lane * 4;
    atomicAdd(ag + 0, ew * hv.x);
    atomicAdd(ag + 1, ew * hv.y);
    atomicAdd(ag + 2, ew * hv.z);
    atomicAdd(ag + 3, ew * hv.w);
  }
}

// ---------------------------------------------------------------------------
// Phase 4: out = elu(agg / max(rowsum, ==0 -> 1)), NaN -> 0
// ---------------------------------------------------------------------------
__global__ void __launch_bounds__(256)
gat_finalize(const float* __restrict__ agg,
             const float* __restrict__ rowsum,
             float* __restrict__ out) {
  const int i = blockIdx.x * blockDim.x + threadIdx.x;
  if (i >= NROWS * DD) return;
  float r = rowsum[i >> 7];
  r = (r == 0.0f) ? 1.0f : r;
  float v = agg[i] / r;
  v = (v != v) ? 0.0f : v;
  out[i] = (v > 0.0f) ? v : (__expf(v) - 1.0f);
}

// ---------------------------------------------------------------------------
// launch
// ---------------------------------------------------------------------------
extern "C" void kernel_launch(void* const* d_in, const int* in_sizes, int n_in,
                              void* d_out, int out_size, void* d_ws, size_t ws_size,
                              hipStream_t stream) {
  const float* x  = (const float*)d_in[0];   // (4,10000,128) f32
  const float* W  = (const float*)d_in[1];   // (128,128) f32
  const float* a  = (const float*)d_in[2];   // (1,256) f32
  const int*   ei = (const int*)d_in[3];     // (2,320000) i32
  float* out = (float*)d_out;                // (4,10000,128) f32

  // workspace layout (floats)
  float* ws     = (float*)d_ws;
  float* h      = ws;                        // 5,120,000
  float* agg    = ws + (size_t)NROWS * DD;   // 5,120,000
  float* fsrc   = agg + (size_t)NROWS * DD;  // 40,000
  float* fdst   = fsrc + NROWS;              // 40,000
  float* rowsum = fdst + NROWS;              // 40,000
  const int nzero = NROWS * DD + 3 * NROWS;  // agg + fsrc + fdst + rowsum

  gat_zero<<<(nzero + 255) / 256, 256, 0, stream>>>(agg, nzero);

  gat_gemm_wmma<<<(NROWS / 16) / 4, 128, 0, stream>>>(x, W, h);      // 625 blocks

  gat_attn_coef<<<NROWS / 8, 256, 0, stream>>>(h, a, fsrc, fdst);    // 5000 blocks

  gat_edge<<<NE / 8, 256, 0, stream>>>(ei, h, fsrc, fdst, agg, rowsum); // 40000 blocks

  gat_finalize<<<(NROWS * DD) / 256, 256, 0, stream>>>(agg, rowsum, out);
}